// GINEConv_Binary_40235253629332
// MI455X (gfx1250) — compile-verified
//
#include <hip/hip_runtime.h>

typedef float v2f __attribute__((ext_vector_type(2)));
typedef float v8f __attribute__((ext_vector_type(8)));

#define NPAD 16

// ---------------- utility kernels ----------------

__global__ void zero_f32(float* __restrict__ p, int n) {
    int i = blockIdx.x * blockDim.x + threadIdx.x;
    if (i < n) p[i] = 0.f;
}

__global__ void pad_nodes(const float* __restrict__ x, float* __restrict__ h, int N) {
    int i = blockIdx.x * blockDim.x + threadIdx.x;
    if (i >= N * NPAD) return;
    int n = i >> 4, k = i & 15;
    h[i] = (k < 12) ? x[n * 12 + k] : 0.f;
}

// Fuse edge-converter linears and build padded/transposed B matrices for WMMA.
// We2f[15][4]  = Wec2 @ Wec1            (so ea2 = edge_attr @ We2f^T + be2f)
// be2f[15]     = Wec2 @ bec1 + bec2
// Bp[k][n]     = W[n][k] zero-padded to 16x16  (B-matrix so D = A x B = in @ W^T)
__global__ void prep(const float* __restrict__ Wec1, const float* __restrict__ bec1,
                     const float* __restrict__ Wec2, const float* __restrict__ bec2,
                     const float* __restrict__ Wl0, const float* __restrict__ Wl1,
                     const float* __restrict__ Wl2,
                     float* __restrict__ We2f, float* __restrict__ be2f,
                     float* __restrict__ Bp0, float* __restrict__ Bp1, float* __restrict__ Bp2) {
    int t = threadIdx.x;
    if (t < 60) {
        int i = t >> 2, j = t & 3;
        float s = 0.f;
        for (int k = 0; k < 12; ++k) s += Wec2[i * 12 + k] * Wec1[k * 4 + j];
        We2f[t] = s;
    }
    if (t < 16) {
        float s = 0.f;
        if (t < 15) {
            s = bec2[t];
            for (int k = 0; k < 12; ++k) s += Wec2[t * 12 + k] * bec1[k];
        }
        be2f[t] = s;
    }
    if (t < 256) {
        int k = t >> 4, n = t & 15;
        Bp0[t] = (k < 12 && n < 15) ? Wl0[n * 12 + k] : 0.f;
        Bp1[t] = (k < 15 && n < 15) ? Wl1[n * 15 + k] : 0.f;
        Bp2[t] = (k < 15 && n < 15) ? Wl2[n * 15 + k] : 0.f;
    }
}

// ---------------- edge scatter (message + aggregate) ----------------
// One thread per edge. Edge feature vector recomputed in-register from the
// 4-wide raw attribute (avoids materializing 150-190 MB ea/ea2 streams).
// msg = relu(h[src] + ev); atomic add into aggr[dst] (aggr fits in L2).
template <int D>
__global__ void edge_scatter(const float* __restrict__ hin, const int* __restrict__ ei,
                             const float* __restrict__ eattr,
                             const float* __restrict__ We, const float* __restrict__ be,
                             float* __restrict__ aggr, int E) {
    int e = blockIdx.x * blockDim.x + threadIdx.x;
    if (e >= E) return;
    int s = ei[e];
    int d = ei[E + e];
    const float4 at = *(const float4*)(eattr + (size_t)e * 4);
    const float4* hp = (const float4*)(hin + (size_t)s * NPAD);
    float xs[16];
    ((float4*)xs)[0] = hp[0];
    ((float4*)xs)[1] = hp[1];
    ((float4*)xs)[2] = hp[2];
    ((float4*)xs)[3] = hp[3];
    float* ap = aggr + (size_t)d * NPAD;
#pragma unroll
    for (int j = 0; j < D; ++j) {
        float ev = be[j] + We[j * 4 + 0] * at.x + We[j * 4 + 1] * at.y +
                   We[j * 4 + 2] * at.z + We[j * 4 + 3] * at.w;
        float m = xs[j] + ev;
        m = m > 0.f ? m : 0.f;  // relu
        atomicAdd(ap + j, m);
    }
}

// ---------------- node linear via WMMA f32 16x16x4 ----------------
// One wave (32 lanes) per 16-node tile. D = (hin+aggr)[16x16] x Bp[16x16],
// K=16 split into four chained V_WMMA_F32_16X16X4_F32. Then +bias, ELU.
__global__ void node_linear_wmma(const float* __restrict__ hin, const float* __restrict__ aggr,
                                 const float* __restrict__ Bp, const float* __restrict__ bias,
                                 float* __restrict__ hout, int ntiles) {
    int tile = blockIdx.x;
    if (tile >= ntiles) return;  // uniform per wave; EXEC stays all-ones
    int lane = threadIdx.x;
    int hi = lane >> 4;
    int m = lane & 15;  // A row / B,C,D column
    int nb = tile * 16;

    v2f a[4], b[4];
#pragma unroll
    for (int kc = 0; kc < 4; ++kc) {
        int k0 = kc * 4 + 2 * hi;
        size_t ro = (size_t)(nb + m) * NPAD;
        a[kc].x = hin[ro + k0] + aggr[ro + k0];
        a[kc].y = hin[ro + k0 + 1] + aggr[ro + k0 + 1];
        b[kc].x = Bp[k0 * 16 + m];
        b[kc].y = Bp[(k0 + 1) * 16 + m];
    }
    v8f c = {};
    c = __builtin_amdgcn_wmma_f32_16x16x4_f32(false, a[0], false, b[0], (short)0, c, false, false);
    c = __builtin_amdgcn_wmma_f32_16x16x4_f32(false, a[1], false, b[1], (short)0, c, false, false);
    c = __builtin_amdgcn_wmma_f32_16x16x4_f32(false, a[2], false, b[2], (short)0, c, false, false);
    c = __builtin_amdgcn_wmma_f32_16x16x4_f32(false, a[3], false, b[3], (short)0, c, false, false);

    float bb = (m < 15) ? bias[m] : 0.f;
#pragma unroll
    for (int i = 0; i < 8; ++i) {
        int row = i + 8 * hi;
        float v = c[i] + bb;
        v = (v > 0.f) ? v : (__expf(v) - 1.f);  // ELU (elu(0)=0 keeps pad col zero)
        hout[(size_t)(nb + row) * NPAD + m] = v;
    }
}

// ---------------- graph pooling ----------------
__global__ void pool_nodes(const float* __restrict__ h, const int* __restrict__ batch,
                           float* __restrict__ pooled, int N) {
    int n = blockIdx.x * blockDim.x + threadIdx.x;
    if (n >= N) return;
    int g = batch[n];
    const float* hp = h + (size_t)n * NPAD;
    float* pp = pooled + (size_t)g * NPAD;
#pragma unroll
    for (int j = 0; j < 15; ++j) atomicAdd(pp + j, hp[j]);
}

// ---------------- output head (tiny) ----------------
__global__ void head(const float* __restrict__ pooled,
                     const float* __restrict__ Wo1, const float* __restrict__ bo1,
                     const float* __restrict__ Wo2, const float* __restrict__ bo2,
                     float* __restrict__ out, int G) {
    int g = blockIdx.x * blockDim.x + threadIdx.x;
    if (g >= G) return;
    const float* p = pooled + (size_t)g * NPAD;
    float t[15];
#pragma unroll
    for (int i = 0; i < 15; ++i) {
        float s = bo1[i];
#pragma unroll
        for (int k = 0; k < 15; ++k) s += Wo1[i * 15 + k] * p[k];
        t[i] = s;
    }
#pragma unroll
    for (int c = 0; c < 2; ++c) {
        float s = bo2[c];
#pragma unroll
        for (int k = 0; k < 15; ++k) s += Wo2[c * 15 + k] * t[k];
        out[g * 2 + c] = s;
    }
}

// ---------------- launch ----------------
extern "C" void kernel_launch(void* const* d_in, const int* in_sizes, int n_in,
                              void* d_out, int out_size, void* d_ws, size_t ws_size,
                              hipStream_t stream) {
    const float* x     = (const float*)d_in[0];
    const int*   ei    = (const int*)d_in[1];
    const float* eattr = (const float*)d_in[2];
    const int*   batch = (const int*)d_in[3];
    const float* Wec1  = (const float*)d_in[4];
    const float* bec1  = (const float*)d_in[5];
    const float* Wec2  = (const float*)d_in[6];
    const float* bec2  = (const float*)d_in[7];
    const float* Wl0   = (const float*)d_in[8];
    const float* bl0   = (const float*)d_in[9];
    const float* Wl1   = (const float*)d_in[10];
    const float* bl1   = (const float*)d_in[11];
    const float* Wl2   = (const float*)d_in[12];
    const float* bl2   = (const float*)d_in[13];
    const float* Wo1   = (const float*)d_in[14];
    const float* bo1   = (const float*)d_in[15];
    const float* Wo2   = (const float*)d_in[16];
    const float* bo2   = (const float*)d_in[17];

    const int N = in_sizes[0] / 12;
    const int E = in_sizes[2] / 4;
    const int G = out_size / 2;
    const int ntiles = (N + 15) / 16;
    const int nrows = ntiles * 16;          // padded node count
    const int nh = nrows * NPAD;            // floats per node buffer

    float* ws = (float*)d_ws;
    size_t off = 0;
    float* hA     = ws + off; off += (size_t)nh;
    float* hB     = ws + off; off += (size_t)nh;
    float* aggr   = ws + off; off += (size_t)nh;
    float* pooled = ws + off; off += (size_t)G * NPAD;
    float* We2f   = ws + off; off += 64;
    float* be2f   = ws + off; off += 16;
    float* Bp0    = ws + off; off += 256;
    float* Bp1    = ws + off; off += 256;
    float* Bp2    = ws + off; off += 256;

    const int TB = 256;
    prep<<<1, 256, 0, stream>>>(Wec1, bec1, Wec2, bec2, Wl0, Wl1, Wl2,
                                We2f, be2f, Bp0, Bp1, Bp2);
    zero_f32<<<(nh + TB - 1) / TB, TB, 0, stream>>>(hA, nh);  // clear tail rows too
    pad_nodes<<<(N * NPAD + TB - 1) / TB, TB, 0, stream>>>(x, hA, N);

    // ---- layer 0: 12 -> 15 ----
    zero_f32<<<(nh + TB - 1) / TB, TB, 0, stream>>>(aggr, nh);
    edge_scatter<12><<<(E + TB - 1) / TB, TB, 0, stream>>>(hA, ei, eattr, Wec1, bec1, aggr, E);
    node_linear_wmma<<<ntiles, 32, 0, stream>>>(hA, aggr, Bp0, bl0, hB, ntiles);

    // ---- layer 1: 15 -> 15 (fused edge weights) ----
    zero_f32<<<(nh + TB - 1) / TB, TB, 0, stream>>>(aggr, nh);
    edge_scatter<15><<<(E + TB - 1) / TB, TB, 0, stream>>>(hB, ei, eattr, We2f, be2f, aggr, E);
    node_linear_wmma<<<ntiles, 32, 0, stream>>>(hB, aggr, Bp1, bl1, hA, ntiles);

    // ---- layer 2: 15 -> 15 ----
    zero_f32<<<(nh + TB - 1) / TB, TB, 0, stream>>>(aggr, nh);
    edge_scatter<15><<<(E + TB - 1) / TB, TB, 0, stream>>>(hA, ei, eattr, We2f, be2f, aggr, E);
    node_linear_wmma<<<ntiles, 32, 0, stream>>>(hA, aggr, Bp2, bl2, hB, ntiles);

    // ---- pool + head ----
    zero_f32<<<(G * NPAD + TB - 1) / TB, TB, 0, stream>>>(pooled, G * NPAD);
    pool_nodes<<<(N + TB - 1) / TB, TB, 0, stream>>>(hB, batch, pooled, N);
    head<<<(G + TB - 1) / TB, TB, 0, stream>>>(pooled, Wo1, bo1, Wo2, bo2, (float*)d_out, G);
}